// HoughVoting_57483842289943
// MI455X (gfx1250) — compile-verified
//
#include <hip/hip_runtime.h>
#include <math.h>

// ---------------- problem constants (from reference) ----------------
#define N_IMG 4
#define H_    480
#define W_    640
#define HW_   (H_ * W_)
#define D_    20           // INLIER_DISTANCE
#define B_    30           // ANGLE_DISCRETIZATION
#define SKIP_ 10           // SKIP_PIXELS
#define KRAD_ 10           // OBJECT_CENTER_KERNEL_RADIUS
#define MAXO_ 16           // MAX_OBJECTS
#define TAB_  (41 * 41)    // (2D+1)^2 offset grid incl. invalid entries
#define CAP_  1024         // candidate list capacity per image
#define PI_F  3.14159265358979323846f

typedef float v2f __attribute__((ext_vector_type(2)));
typedef float v8f __attribute__((ext_vector_type(8)));
typedef int   gv2i __attribute__((vector_size(8)));   // matches async b64 builtin param
typedef unsigned long long u64;

// ---------------- gfx1250 feature probes (compile-safe) ----------------
#if defined(__has_builtin)
#  if __has_builtin(__builtin_amdgcn_global_load_async_to_lds_b64) && \
      __has_builtin(__builtin_amdgcn_global_load_async_to_lds_b32) && \
      __has_builtin(__builtin_amdgcn_s_wait_asynccnt)
#    define HAVE_ASYNC_LDS 1
#  endif
#  if __has_builtin(__builtin_amdgcn_wmma_f32_16x16x4_f32)
#    define HAVE_WMMA_F32 1
#  endif
#endif

// ---------------- workspace layout (bytes) ----------------
// mask   : u32 [N*HW]    per-pixel 30-bit angle-bin vote mask
// tmp    : u8  [N*HW]    row-max of popcount (separable max-pool pass 1)
// uvtab  : float2[TAB_]  unit offset vectors (uy,ux); 0 for invalid
// metatab: u32 [TAB_]    (oy+20) | (ox+20)<<8 | bin<<16 ; 0xFFFFFFFF invalid
// ccnt   : u32 [N]       candidate counts
// keys   : u64 [N*CAP_]  (count<<32) | ~idx   (sorts val desc, idx asc)
// cenf   : f32 [N*16*2]  (cy,cx) per slot as float
// validm : u32 [N]       16-bit valid mask per image
static constexpr size_t WS_MASK = 0;
static constexpr size_t WS_TMP  = WS_MASK + (size_t)N_IMG * HW_ * 4;
static constexpr size_t WS_UV   = WS_TMP  + (size_t)N_IMG * HW_;       // 8B aligned
static constexpr size_t WS_META = WS_UV   + (size_t)TAB_ * 8;
static constexpr size_t WS_CCNT = WS_META + ((size_t)TAB_ * 4 + 15 & ~(size_t)15);
static constexpr size_t WS_KEYS = WS_CCNT + 16;
static constexpr size_t WS_CENF = WS_KEYS + (size_t)N_IMG * CAP_ * 8;
static constexpr size_t WS_VAL  = WS_CENF + (size_t)N_IMG * MAXO_ * 2 * 4;

// ---------------- kernel 0: zero vote masks + candidate counters ----------------
__global__ void __launch_bounds__(256) k_init(unsigned* mask, unsigned* ccnt, int total) {
    int i = blockIdx.x * 256 + threadIdx.x;
    if (i < total) mask[i] = 0u;
    if (blockIdx.x == 0 && threadIdx.x < N_IMG) ccnt[threadIdx.x] = 0u;
}

// ---------------- kernel 1: build offset table ----------------
__global__ void __launch_bounds__(256) k_table(float2* uvtab, unsigned* metatab) {
    int i = blockIdx.x * 256 + threadIdx.x;
    if (i >= TAB_) return;
    int yy = i / 41 - D_;
    int xx = i % 41 - D_;
    int r2 = yy * yy + xx * xx;
    float2 uv; unsigned meta;
    if (r2 > 0 && r2 <= D_ * D_) {
        float d = sqrtf((float)r2);
        uv.x = (float)yy / d;
        uv.y = (float)xx / d;
        float ang = atan2f((float)yy, (float)xx);
        int b = (int)((ang + PI_F) * ((float)B_ / (2.0f * PI_F)));
        b = b < 0 ? 0 : (b > B_ - 1 ? B_ - 1 : b);
        meta = (unsigned)(yy + D_) | ((unsigned)(xx + D_) << 8) | ((unsigned)b << 16);
    } else {
        uv.x = 0.0f; uv.y = 0.0f;
        meta = 0xFFFFFFFFu;
    }
    uvtab[i] = uv;
    metatab[i] = meta;
}

// ---------------- kernel 2: vote scatter (offset table staged via async->LDS) ----------------
__global__ void __launch_bounds__(256) k_vote(const int* __restrict__ label,
                                              const float* __restrict__ dirs,
                                              const float2* __restrict__ uvtab,
                                              const unsigned* __restrict__ metatab,
                                              unsigned* __restrict__ mask) {
    __shared__ float2   sUV[TAB_];
    __shared__ unsigned sMeta[TAB_];
#if defined(HAVE_ASYNC_LDS)
    for (int i = threadIdx.x; i < TAB_; i += 256) {
        __builtin_amdgcn_global_load_async_to_lds_b64(
            (gv2i*)(uvtab + i), (gv2i*)(sUV + i), 0, 0);
        __builtin_amdgcn_global_load_async_to_lds_b32(
            (int*)(metatab + i), (int*)(sMeta + i), 0, 0);
    }
    __builtin_amdgcn_s_wait_asynccnt(0);
#else
    for (int i = threadIdx.x; i < TAB_; i += 256) {
        sUV[i]   = uvtab[i];
        sMeta[i] = metatab[i];
    }
#endif
    __syncthreads();

    const int VOT = HW_ / SKIP_;
    int t = blockIdx.x * 256 + threadIdx.x;
    if (t >= N_IMG * VOT) return;
    int n = t / VOT;
    int p = (t - n * VOT) * SKIP_;
    if (label[(size_t)n * HW_ + p] <= 0) return;

    int y = p / W_, x = p - y * W_;
    float dy = dirs[(size_t)n * 2 * HW_ + p];
    float dx = dirs[(size_t)n * 2 * HW_ + HW_ + p];
    float nrm = sqrtf(dy * dy + dx * dx) + 1e-12f;
    float nyv = dy / nrm, nxv = dx / nrm;
    unsigned* mimg = mask + (size_t)n * HW_;

    for (int i = 0; i < TAB_; ++i) {
        unsigned meta = sMeta[i];
        float2 uv = sUV[i];
        if (meta == 0xFFFFFFFFu) continue;
        if (nyv * uv.x + nxv * uv.y >= 0.9f) {
            int ty = y + (int)(meta & 0xFFu) - D_;
            int tx = x + (int)((meta >> 8) & 0xFFu) - D_;
            unsigned b = (meta >> 16) & 0xFFu;
            if ((unsigned)ty < (unsigned)H_ && (unsigned)tx < (unsigned)W_)
                atomicOr(&mimg[ty * W_ + tx], 1u << b);
        }
    }
}

// ---------------- kernel 3: separable max-pool pass 1 (row max of popcount) ----------------
__global__ void __launch_bounds__(256) k_rowmax(const unsigned* __restrict__ mask,
                                                unsigned char* __restrict__ tmp) {
    int i = blockIdx.x * 256 + threadIdx.x;
    if (i >= N_IMG * HW_) return;
    int n = i / HW_;
    int p = i - n * HW_;
    int y = p / W_, x = p - y * W_;
    const unsigned* row = mask + (size_t)n * HW_ + (size_t)y * W_;
    int x0 = x - KRAD_ < 0 ? 0 : x - KRAD_;
    int x1 = x + KRAD_ > W_ - 1 ? W_ - 1 : x + KRAD_;
    int mv = 0;
    for (int xx = x0; xx <= x1; ++xx) {
        int c = __popc(row[xx]);
        mv = c > mv ? c : mv;
    }
    tmp[i] = (unsigned char)mv;
}

// ---------------- kernel 4: pass 2 (col max) fused with candidate extraction ----------------
__global__ void __launch_bounds__(256) k_colcand(const unsigned* __restrict__ mask,
                                                 const unsigned char* __restrict__ tmp,
                                                 unsigned* __restrict__ ccnt,
                                                 u64* __restrict__ keys) {
    int i = blockIdx.x * 256 + threadIdx.x;
    if (i >= N_IMG * HW_) return;
    int n = i / HW_;
    int p = i - n * HW_;
    int y = p / W_, x = p - y * W_;
    int c = __popc(mask[i]);
    if (c < 15) return;                     // PERCENTAGE_THRESHOLD * B = 15
    int y0 = y - KRAD_ < 0 ? 0 : y - KRAD_;
    int y1 = y + KRAD_ > H_ - 1 ? H_ - 1 : y + KRAD_;
    int pool = 0;
    const unsigned char* col = tmp + (size_t)n * HW_ + x;
    for (int yy = y0; yy <= y1; ++yy) {
        int v = col[(size_t)yy * W_];
        pool = v > pool ? v : pool;
    }
    if (c >= pool) {
        unsigned pos = atomicAdd(&ccnt[n], 1u);
        if (pos < CAP_)
            keys[(size_t)n * CAP_ + pos] =
                ((u64)(unsigned)c << 32) | (u64)(0xFFFFFFFFu - (unsigned)p);
    }
}

// ---------------- kernel 5: per-image top-16 selection (value desc, idx asc) ----------------
__global__ void __launch_bounds__(256) k_topk(const unsigned* __restrict__ ccnt,
                                              const u64* __restrict__ keys,
                                              float* __restrict__ cenf,
                                              unsigned* __restrict__ validm,
                                              float* __restrict__ out_tail) {
    __shared__ u64 skeys[CAP_];
    __shared__ u64 smax[256];
    __shared__ u64 ssel[MAXO_];
    int n = blockIdx.x;
    unsigned cnt = ccnt[n];
    if (cnt > CAP_) cnt = CAP_;
    for (int i = threadIdx.x; i < CAP_; i += 256)
        skeys[i] = (i < (int)cnt) ? keys[(size_t)n * CAP_ + i] : 0ull;
    __syncthreads();

    u64 prev = ~0ull;
    for (int r = 0; r < MAXO_; ++r) {
        u64 mloc = 0;
        for (int i = threadIdx.x; i < (int)cnt; i += 256) {
            u64 k = skeys[i];
            if (k < prev && k > mloc) mloc = k;
        }
        smax[threadIdx.x] = mloc;
        __syncthreads();
        if (threadIdx.x == 0) {
            u64 best = 0;
            for (int i = 0; i < 256; ++i)
                if (smax[i] > best) best = smax[i];
            ssel[r] = best;                 // 0 => no more candidates
        }
        __syncthreads();
        prev = ssel[r];
    }

    if (threadIdx.x == 0) {
        int idxs[MAXO_];
        int num = 0;
        unsigned vbits = 0;
        for (int s = 0; s < MAXO_; ++s) {
            if (ssel[s] != 0ull) {
                idxs[s] = (int)(0xFFFFFFFFu - (unsigned)(ssel[s] & 0xFFFFFFFFu));
                vbits |= 1u << s;
                ++num;
            } else {
                idxs[s] = -1;
            }
        }
        // fill invalid slots with the smallest zero-score indices (top_k tie behavior)
        int fp = 0;
        for (int s = 0; s < MAXO_; ++s) {
            if (idxs[s] >= 0) continue;
            for (;;) {
                bool used = false;
                for (int t = 0; t < MAXO_; ++t)
                    if (idxs[t] == fp) { used = true; break; }
                if (!used) break;
                ++fp;
            }
            idxs[s] = fp;
            ++fp;
        }
        validm[n] = vbits;
        out_tail[n] = (float)num;           // num_objects
        for (int s = 0; s < MAXO_; ++s) {
            int cy = idxs[s] / W_, cx = idxs[s] % W_;
            out_tail[N_IMG + (n * MAXO_ + s) * 2 + 0] = (float)cy;
            out_tail[N_IMG + (n * MAXO_ + s) * 2 + 1] = (float)cx;
            cenf[(n * MAXO_ + s) * 2 + 0] = (float)cy;
            cenf[(n * MAXO_ + s) * 2 + 1] = (float)cx;
        }
    }
}

// ---------------- kernel 6: pixel->object assignment via WMMA f32 16x16x4 ----------------
// numerator(pixel p, object m) = ny*(cy-py) + nx*(cx-px)
//   = [cy, cx, 1, 0] . [ny, nx, -(ny*py+nx*px), 0]  -> A(16 obj x 4) * B(4 x 16 px)
__global__ void __launch_bounds__(256) k_assign(const int* __restrict__ label,
                                                const float* __restrict__ dirs,
                                                const float* __restrict__ cenf,
                                                const unsigned* __restrict__ validm,
                                                float* __restrict__ out) {
    __shared__ float sCy[MAXO_], sCx[MAXO_];
    __shared__ unsigned sValid;
    int n = blockIdx.y;
    if (threadIdx.x < MAXO_) {
        sCy[threadIdx.x] = cenf[(n * MAXO_ + threadIdx.x) * 2 + 0];
        sCx[threadIdx.x] = cenf[(n * MAXO_ + threadIdx.x) * 2 + 1];
    }
    if (threadIdx.x == 0) sValid = validm[n];
    __syncthreads();

    int wave = threadIdx.x >> 5;
    int lane = threadIdx.x & 31;
    int g  = blockIdx.x * 8 + wave;        // 19200 groups of 16 pixels per image
    int lp = g * 16 + (lane & 15);         // pixel handled by this lane (pair)
    float py = (float)(lp / W_);
    float px = (float)(lp % W_);
    float dy = dirs[(size_t)n * 2 * HW_ + lp];
    float dx = dirs[(size_t)n * 2 * HW_ + HW_ + lp];
    float nrm = sqrtf(dy * dy + dx * dx) + 1e-12f;
    float nyv = dy / nrm, nxv = dx / nrm;
    int m = lane & 15;

    v8f acc;
#if defined(HAVE_WMMA_F32)
    // A (16x4 f32): lanes 0-15 hold K0,K1 for M=lane; lanes 16-31 hold K2,K3.
    // B (4x16 f32): lanes 0-15 hold K0,K1 for N=lane; lanes 16-31 hold K2,K3.
    v2f A, Bv;
    if (lane < 16) { A[0] = sCy[m];  A[1] = sCx[m]; }
    else           { A[0] = 1.0f;    A[1] = 0.0f;   }
    if (lane < 16) { Bv[0] = nyv;    Bv[1] = nxv;   }
    else           { Bv[0] = -(nyv * py + nxv * px); Bv[1] = 0.0f; }
    v8f C = {0.f, 0.f, 0.f, 0.f, 0.f, 0.f, 0.f, 0.f};
    acc = __builtin_amdgcn_wmma_f32_16x16x4_f32(false, A, false, Bv, (short)0, C,
                                                false, false);
#else
    int obS = (lane < 16) ? 0 : 8;
    for (int v = 0; v < 8; ++v)
        acc[v] = nyv * (sCy[obS + v] - py) + nxv * (sCx[obS + v] - px);
#endif

    // D layout: lane L<16 -> pixel N=L, objects M=0..7 in acc[0..7];
    //           lane L+16 -> pixel N=L, objects M=8..15.
    int obase = (lane < 16) ? 0 : 8;
    float bestc = -__builtin_inff();
    int   besti = 0;
    for (int v = 0; v < 8; ++v) {
        int mm = obase + v;
        float vy = sCy[mm] - py, vx = sCx[mm] - px;
        float dist = sqrtf(vy * vy + vx * vx) + 1e-12f;
        float c = acc[v] / dist;
        bool ok = (sValid >> mm) & 1u;
        c = ok ? c : -__builtin_inff();
        if (c > bestc) { bestc = c; besti = mm; }
    }
    // merge the two object halves; lower half holds lower indices -> strict > keeps ties left
    float oc = __shfl_xor(bestc, 16, 32);
    int   oi = __shfl_xor(besti, 16, 32);
    if (oc > bestc) { bestc = oc; besti = oi; }

    if (lane < 16) {
        bool fg = label[(size_t)n * HW_ + lp] > 0;
        out[(size_t)n * HW_ + lp] = (fg && bestc >= 0.9f) ? (float)(besti + 2) : 0.0f;
    }
}

// ---------------- host launcher ----------------
extern "C" void kernel_launch(void* const* d_in, const int* in_sizes, int n_in,
                              void* d_out, int out_size, void* d_ws, size_t ws_size,
                              hipStream_t stream) {
    (void)in_sizes; (void)n_in; (void)out_size; (void)ws_size;
    const int*   label = (const int*)d_in[0];
    const float* dirs  = (const float*)d_in[1];
    float*       out   = (float*)d_out;
    char*        ws    = (char*)d_ws;

    unsigned*      mask    = (unsigned*)(ws + WS_MASK);
    unsigned char* tmp     = (unsigned char*)(ws + WS_TMP);
    float2*        uvtab   = (float2*)(ws + WS_UV);
    unsigned*      metatab = (unsigned*)(ws + WS_META);
    unsigned*      ccnt    = (unsigned*)(ws + WS_CCNT);
    u64*           keys    = (u64*)(ws + WS_KEYS);
    float*         cenf    = (float*)(ws + WS_CENF);
    unsigned*      validm  = (unsigned*)(ws + WS_VAL);

    const int totalPix = N_IMG * HW_;                        // 1,228,800
    k_init   <<<totalPix / 256, 256, 0, stream>>>(mask, ccnt, totalPix);
    k_table  <<<(TAB_ + 255) / 256, 256, 0, stream>>>(uvtab, metatab);
    k_vote   <<<(N_IMG * (HW_ / SKIP_)) / 256, 256, 0, stream>>>(label, dirs, uvtab,
                                                                 metatab, mask);
    k_rowmax <<<totalPix / 256, 256, 0, stream>>>(mask, tmp);
    k_colcand<<<totalPix / 256, 256, 0, stream>>>(mask, tmp, ccnt, keys);
    k_topk   <<<N_IMG, 256, 0, stream>>>(ccnt, keys, cenf, validm,
                                         out + (size_t)N_IMG * HW_);
    dim3 ag(HW_ / (16 * 8), N_IMG);                          // 2400 x 4 blocks
    k_assign <<<ag, 256, 0, stream>>>(label, dirs, cenf, validm, out);
}